// MultiBoxLoss_20675972563168
// MI455X (gfx1250) — compile-verified
//
#include <hip/hip_runtime.h>
#include <hip/hip_bf16.h>
#include <stdint.h>

// ---------------------------------------------------------------------------
// SSD MultiBox loss, MI455X (gfx1250, wave32).
// Memory-bound problem (~59 MB traffic -> ~2.5us @ 23.3 TB/s). Design:
//  - conf_data (47MB) streamed once via CDNA5 async-to-LDS (coalesced b128
//    bursts), consumed from LDS with conflict-free stride-21 rows.
//  - per-truth argmax over 8732 priors via hierarchical packed-u64 atomicMax.
//  - hard-negative mining reduced to "sum of top-num_neg of loss_c_mine":
//    per-batch bitonic sort fully inside LDS (64KB of the 320KB/WGP).
//  - block reductions fed through V_WMMA_F32_16X16X4_F32 (ones-matrix
//    product = 16-way f32-exact sum on the matrix pipe).
// ---------------------------------------------------------------------------

#define NUM_B 64
#define NUM_P 8732
#define NUM_C 21
#define NUM_T 16
#define TPB   256
#define SORT_N 16384

typedef float v2f __attribute__((ext_vector_type(2)));
typedef float v8f __attribute__((ext_vector_type(8)));

// ---- wave32 sum via V_WMMA_F32_16X16X4_F32 -------------------------------
// A(16x4): VGPR0 = per-lane value, VGPR1 = 0.  B(4x16) = ones.
// D[m][n] = v[m] + v[m+16] for every n.  Lanes 0-15 sum rows 0-7 of D,
// lanes 16-31 sum rows 8-15; total = lane0 + lane16.
__device__ __forceinline__ float wave_sum_wmma(float v) {
  v2f a; a[0] = v;    a[1] = 0.0f;
  v2f b; b[0] = 1.0f; b[1] = 1.0f;
  v8f c = {};
  v8f d = __builtin_amdgcn_wmma_f32_16x16x4_f32(
      /*neg_a=*/false, a, /*neg_b=*/false, b,
      /*c_mod=*/(short)0, c, /*reuse_a=*/false, /*reuse_b=*/false);
  float acc = d[0] + d[1] + d[2] + d[3] + d[4] + d[5] + d[6] + d[7];
  return __shfl(acc, 0, 32) + __shfl(acc, 16, 32);
}

// Block-wide sum (result valid in thread 0). Must be called with EXEC all-1s
// (unconditionally by every thread in a full block) -- WMMA requirement.
__device__ __forceinline__ float block_sum_wmma(float v, float* sred) {
  const int tid  = threadIdx.x;
  const int lane = tid & 31;
  const int wid  = tid >> 5;
  float w = wave_sum_wmma(v);
  __syncthreads();                 // protect sred reuse across calls
  if (lane == 0) sred[wid] = w;
  __syncthreads();
  float r = 0.0f;
  const int nw = blockDim.x >> 5;
  if (tid < 32) {
    if (tid < nw) r = sred[tid];
    #pragma unroll
    for (int o = 16; o; o >>= 1) r += __shfl_down(r, o, 32);
  }
  return r;
}

__device__ __forceinline__ unsigned long long pack_key(float iou, unsigned p) {
  // iou >= 0 -> float bits are monotone; ~p makes max() prefer smallest p
  // (matches jnp.argmax first-occurrence tie-break).
  return ((unsigned long long)__float_as_uint(iou) << 32) |
         (unsigned long long)(0xFFFFFFFFu - p);
}

// ---------------------------------------------------------------------------
__global__ void init_kernel(unsigned long long* bp, int* npos, float* acc) {
  const int t = threadIdx.x;            // 1024 threads
  if (t < NUM_B * NUM_T) bp[t] = 0ull;  // 1024 entries
  if (t < NUM_B) npos[t] = 0;
  if (t < 3) acc[t] = 0.0f;
}

// ---- kernel 1: match -------------------------------------------------------
__global__ __launch_bounds__(TPB) void match_kernel(
    const float* __restrict__ priors, const float* __restrict__ targets,
    float* __restrict__ bto, int* __restrict__ bti,
    unsigned long long* __restrict__ bp) {
  __shared__ float sTr[NUM_T * 5];
  __shared__ unsigned long long sBest[NUM_T];
  const int b   = blockIdx.y;
  const int p0  = blockIdx.x * TPB;
  const int tid = threadIdx.x;
  const int p   = p0 + tid;

  if (tid < NUM_T * 5) sTr[tid] = targets[b * NUM_T * 5 + tid];
  if (tid < NUM_T)     sBest[tid] = 0ull;
  __syncthreads();

  if (p < NUM_P) {
    const float4 pr = reinterpret_cast<const float4*>(priors)[p];
    const float px1 = pr.x - 0.5f * pr.z, py1 = pr.y - 0.5f * pr.w;
    const float px2 = pr.x + 0.5f * pr.z, py2 = pr.y + 0.5f * pr.w;
    const float areaP = pr.z * pr.w;
    float bestV = -1.0f; int bestJ = 0;
    #pragma unroll 4
    for (int j = 0; j < NUM_T; j++) {
      const float x1 = sTr[j*5], y1 = sTr[j*5+1], x2 = sTr[j*5+2], y2 = sTr[j*5+3];
      float iw = fminf(px2, x2) - fmaxf(px1, x1); iw = fmaxf(iw, 0.0f);
      float ih = fminf(py2, y2) - fmaxf(py1, y1); ih = fmaxf(ih, 0.0f);
      const float inter = iw * ih;
      const float areaT = (x2 - x1) * (y2 - y1);
      const float iou = inter / (areaT + areaP - inter);
      if (iou > bestV) { bestV = iou; bestJ = j; }
      atomicMax(&sBest[j], pack_key(iou, (unsigned)p));   // LDS pre-reduce
    }
    const size_t ix = (size_t)b * NUM_P + p;
    bto[ix] = bestV;
    bti[ix] = bestJ;
  }
  __syncthreads();
  if (tid < NUM_T) atomicMax(&bp[b * NUM_T + tid], sBest[tid]); // 1 atomic/blk/j
}

// ---- kernel 2: best-prior override (ordered scatter) ----------------------
__global__ void override_kernel(const unsigned long long* __restrict__ bp,
                                float* __restrict__ bto, int* __restrict__ bti) {
  const int b = threadIdx.x;
  if (b < NUM_B) {
    for (int j = 0; j < NUM_T; j++) {   // ascending j: later j wins, as in ref
      const unsigned pidx =
          0xFFFFFFFFu - (unsigned)(bp[b * NUM_T + j] & 0xFFFFFFFFull);
      const size_t ix = (size_t)b * NUM_P + pidx;
      bto[ix] = 2.0f;
      bti[ix] = j;
    }
  }
}

// ---- kernel 3: per-prior losses (async-to-LDS streaming + WMMA reduce) ----
__global__ __launch_bounds__(TPB) void per_prior_kernel(
    const float* __restrict__ loc, const float* __restrict__ conf,
    const float* __restrict__ priors, const float* __restrict__ targets,
    const float* __restrict__ bto, const int* __restrict__ bti,
    float* __restrict__ mine, int* __restrict__ npos, float* __restrict__ acc) {
  __shared__ __align__(16) float shConf[TPB * NUM_C];   // 21504 B
  __shared__ float sTr[NUM_T * 5];
  __shared__ float sRed[TPB / 32];

  const int b   = blockIdx.y;
  const int p0  = blockIdx.x * TPB;
  const int tid = threadIdx.x;
  const int valid = min(TPB, NUM_P - p0);   // 256 or 28; both %4==0

  if (tid < NUM_T * 5) sTr[tid] = targets[b * NUM_T * 5 + tid];

  // Stage valid*21 floats of conf_data via CDNA5 async-to-LDS b128 bursts.
  // Global base is uniform per block (SGPR pair); per-lane 32-bit voffset.
  const float* gbase = conf + ((size_t)b * NUM_P + p0) * NUM_C;  // 16B aligned
  const int nchunks = (valid * NUM_C) >> 2;                      // float4 chunks
  for (int c = tid; c < nchunks; c += TPB) {
    unsigned ldsOff = (unsigned)(uintptr_t)(&shConf[c * 4]);     // low32 = LDS addr
    unsigned gOff   = (unsigned)(c * 16);
    asm volatile("global_load_async_to_lds_b128 %0, %1, %2"
                 :: "v"(ldsOff), "v"(gOff), "s"(gbase) : "memory");
  }
  asm volatile("s_wait_asynccnt 0" ::: "memory");
  __syncthreads();

  float lsum = 0.0f, plc = 0.0f, cnt = 0.0f;
  if (tid < valid) {
    const int p = p0 + tid;
    const size_t ix = (size_t)b * NUM_P + p;
    const float over = bto[ix];
    const int   j    = bti[ix];
    const int   ci   = (over < 0.5f) ? 0 : ((int)sTr[j * 5 + 4] + 1);
    const bool  pos  = ci > 0;

    // logsumexp over 21 classes; stride-21 LDS rows are bank-conflict-free.
    const float* row = &shConf[tid * NUM_C];
    float m = row[0];
    #pragma unroll
    for (int k = 1; k < NUM_C; k++) m = fmaxf(m, row[k]);
    float se = 0.0f;
    #pragma unroll
    for (int k = 0; k < NUM_C; k++) se += __expf(row[k] - m);
    const float lse = m + __logf(se);
    const float lc  = lse - row[ci];
    mine[ix] = pos ? 0.0f : lc;      // hard-negative mining key

    if (pos) {
      plc = lc; cnt = 1.0f;
      const float4 pr = reinterpret_cast<const float4*>(priors)[p];
      const float x1 = sTr[j*5], y1 = sTr[j*5+1], x2 = sTr[j*5+2], y2 = sTr[j*5+3];
      const float g0 = (0.5f * (x1 + x2) - pr.x) / (0.1f * pr.z);
      const float g1 = (0.5f * (y1 + y2) - pr.y) / (0.1f * pr.w);
      const float g2 = __logf((x2 - x1) / pr.z) * 5.0f;   // /VARIANCES[1]
      const float g3 = __logf((y2 - y1) / pr.w) * 5.0f;
      const float4 ld = reinterpret_cast<const float4*>(loc)[ix];
      const float d0 = fabsf(ld.x - g0), d1 = fabsf(ld.y - g1);
      const float d2 = fabsf(ld.z - g2), d3 = fabsf(ld.w - g3);
      lsum = (d0 < 1.0f ? 0.5f * d0 * d0 : d0 - 0.5f)
           + (d1 < 1.0f ? 0.5f * d1 * d1 : d1 - 0.5f)
           + (d2 < 1.0f ? 0.5f * d2 * d2 : d2 - 0.5f)
           + (d3 < 1.0f ? 0.5f * d3 * d3 : d3 - 0.5f);
    }
  }
  // Reconverged here: EXEC all-1s for the WMMA reductions.
  const float rl = block_sum_wmma(lsum, sRed);
  const float rp = block_sum_wmma(plc,  sRed);
  const float rc = block_sum_wmma(cnt,  sRed);
  if (tid == 0) {
    if (rl != 0.0f) atomicAdd(&acc[0], rl);
    if (rp != 0.0f) atomicAdd(&acc[1], rp);
    const int ip = (int)(rc + 0.5f);
    if (ip) atomicAdd(&npos[b], ip);
  }
}

// ---- kernel 4: per-batch bitonic sort in LDS + top-num_neg sum ------------
__global__ __launch_bounds__(1024) void sort_topk_kernel(
    const float* __restrict__ mine, const int* __restrict__ npos,
    float* __restrict__ acc) {
  __shared__ float s[SORT_N];                 // 64 KB of 320 KB/WGP
  const int b = blockIdx.x, tid = threadIdx.x;

  for (int i = tid; i < SORT_N; i += 1024)
    s[i] = (i < NUM_P) ? mine[(size_t)b * NUM_P + i] : -1.0f;  // pad sinks last
  __syncthreads();

  for (int k = 2; k <= SORT_N; k <<= 1) {
    for (int j = k >> 1; j > 0; j >>= 1) {
      for (int t = tid; t < SORT_N; t += 1024) {
        const int ixj = t ^ j;
        if (ixj > t) {
          const bool desc = ((t & k) == 0);     // overall descending sort
          const float a = s[t], c = s[ixj];
          if (desc ? (a < c) : (a > c)) { s[t] = c; s[ixj] = a; }
        }
      }
      __syncthreads();
    }
  }

  const int np = npos[b];
  const int nn = min(3 * np, NUM_P - 1);        // num_neg
  float part = 0.0f;
  for (int i = tid; i < nn; i += 1024) part += s[i];
  #pragma unroll
  for (int o = 16; o; o >>= 1) part += __shfl_down(part, o, 32);
  __syncthreads();                              // done reading sorted data
  if ((tid & 31) == 0) s[tid >> 5] = part;      // 32 waves -> s[0..31]
  __syncthreads();
  if (tid < 32) {
    float r = s[tid];
    #pragma unroll
    for (int o = 16; o; o >>= 1) r += __shfl_down(r, o, 32);
    if (tid == 0) atomicAdd(&acc[2], r);
  }
}

// ---- kernel 5: finalize ----------------------------------------------------
__global__ void finalize_kernel(const int* __restrict__ npos,
                                const float* __restrict__ acc,
                                float* __restrict__ out) {
  __shared__ int sN[NUM_B];
  const int tid = threadIdx.x;
  sN[tid] = npos[tid];
  __syncthreads();
  if (tid == 0) {
    int N = 0;
    for (int i = 0; i < NUM_B; i++) N += sN[i];
    const float fN = (float)N;
    out[0] = acc[0] / fN;                 // loss_l / N
    out[1] = (acc[1] + acc[2]) / fN;      // (pos loss_c + hard-neg loss_c) / N
  }
}

// ---------------------------------------------------------------------------
extern "C" void kernel_launch(void* const* d_in, const int* in_sizes, int n_in,
                              void* d_out, int out_size, void* d_ws, size_t ws_size,
                              hipStream_t stream) {
  const float* loc     = (const float*)d_in[0];   // (64, 8732, 4)
  const float* conf    = (const float*)d_in[1];   // (64, 8732, 21)
  const float* priors  = (const float*)d_in[2];   // (8732, 4)
  const float* targets = (const float*)d_in[3];   // (64, 16, 5)
  float* out = (float*)d_out;                     // 2 floats

  // Workspace layout (~6.7 MB): lives in the 192 MB L2 between kernels.
  char* ws = (char*)d_ws;
  const size_t BP = (size_t)NUM_B * NUM_P;
  unsigned long long* bp = (unsigned long long*)ws;          // 8192 B
  float* bto  = (float*)(ws + 8192);                         // BP * 4
  int*   bti  = (int*)  (ws + 8192 + BP * 4);                // BP * 4
  float* mine = (float*)(ws + 8192 + BP * 8);                // BP * 4
  int*   npos = (int*)  (ws + 8192 + BP * 12);               // 256 B
  float* acc  = (float*)(ws + 8192 + BP * 12 + 256);         // 3 floats

  const dim3 grid((NUM_P + TPB - 1) / TPB, NUM_B);           // (35, 64)

  init_kernel<<<1, 1024, 0, stream>>>(bp, npos, acc);
  match_kernel<<<grid, TPB, 0, stream>>>(priors, targets, bto, bti, bp);
  override_kernel<<<1, 64, 0, stream>>>(bp, bto, bti);
  per_prior_kernel<<<grid, TPB, 0, stream>>>(loc, conf, priors, targets,
                                             bto, bti, mine, npos, acc);
  sort_topk_kernel<<<NUM_B, 1024, 0, stream>>>(mine, npos, acc);
  finalize_kernel<<<1, NUM_B, 0, stream>>>(npos, acc, out);
}